// FastGuidedFilter_29892972380250
// MI455X (gfx1250) — compile-verified
//
#include <hip/hip_runtime.h>
#include <stdint.h>

// Guided filter, fully fused single-pass tile kernel for gfx1250 (MI455X).
// B=16, C=3, H=W=1024, fp32, radius=5, eps=1e-3.

#define RADIUS 5
#define K11    (2 * RADIUS + 1)          // 11
#define EPSF   0.001f

#define BATCH  16
#define CHANS  3
#define IMH    1024
#define IMW    1024

#define TW 64                            // output tile width
#define TH 48                            // output tile height
#define SW (TW + 4 * RADIUS)             // 84  : I/p halo width  (2r each side)
#define SH (TH + 4 * RADIUS)             // 68  : I/p halo height
#define AW (TW + 2 * RADIUS)             // 74  : a/b region width (r each side)
#define AH (TH + 2 * RADIUS)             // 58  : a/b region height

// ---- LDS layout (floats), with buffer reuse across phases ----
#define OFF_SI   0                       // sI   : SH*SW = 5712   (live whole kernel)
#define OFF_SP   (SH * SW)               // sP   : 5712           (dead after phase B)
#define OFF_AT   OFF_SP                  // aT   : AH*AW = 4292   (reuses sP)
#define OFF_HI   (2 * SH * SW)           // hI   : SH*AW = 5032
#define OFF_HA   OFF_HI                  // ha   : AH*TW = 3712   (reuses hI)
#define OFF_HP   (OFF_HI + SH * AW)      // hP   : 5032
#define OFF_HB   OFF_HP                  // hb   : 3712           (reuses hP)
#define OFF_HIP  (OFF_HP + SH * AW)      // hIp  : 5032
#define OFF_QT   OFF_HIP                 // qT   : TH*TW = 3072   (reuses hIp)
#define OFF_HII  (OFF_HIP + SH * AW)     // hII  : 5032
#define OFF_BT   (OFF_HII + SH * AW)     // bT   : 4292
#define LDS_FLOATS (OFF_BT + AH * AW)    // 35844 floats = 143,376 bytes

// ---- gfx1250 async global->LDS copy (ASYNCcnt path), with safe fallback ----
#if defined(__gfx1250__) && __has_builtin(__builtin_amdgcn_global_load_async_to_lds_b32)
#define HAVE_ASYNC 1
#else
#define HAVE_ASYNC 0
#endif

__device__ __forceinline__ void async_cp_b32(const float* g, float* l) {
#if HAVE_ASYNC
  // Builtin parameter types (from hipcc diagnostic): AS1 int*, AS3 int*.
  // AS0->AS1 is a bit-identity cast; an AS0 LDS pointer's low 32 bits are the
  // LDS offset (shared aperture lives in the high 32 bits), so the integer
  // detour produces valid addrspace-qualified pointers on AMDGPU.
  __builtin_amdgcn_global_load_async_to_lds_b32(
      (__attribute__((address_space(1))) int*)(uintptr_t)(const void*)g,
      (__attribute__((address_space(3))) int*)(uint32_t)(uintptr_t)(void*)l,
      0, 0);
#else
  *l = *g;
#endif
}

__device__ __forceinline__ void async_wait_all() {
#if HAVE_ASYNC
#if __has_builtin(__builtin_amdgcn_s_wait_asynccnt)
  __builtin_amdgcn_s_wait_asynccnt(0);
#else
  asm volatile("s_wait_asynccnt 0" ::: "memory");
#endif
#endif
}

__device__ __forceinline__ int imin(int a, int b) { return a < b ? a : b; }

__global__ __launch_bounds__(256, 1)
void guided_filter_fused(const float* __restrict__ I,
                         const float* __restrict__ P,
                         float* __restrict__ Q) {
  __shared__ float lds[LDS_FLOATS];

  const int tid   = threadIdx.x;
  const int plane = blockIdx.z;                      // b*C + c
  const int ox0   = blockIdx.x * TW;
  const int oy0   = blockIdx.y * TH;
  const long planeOff = (long)plane * IMH * IMW;
  const float* gI = I + planeOff;
  const float* gP = P + planeOff;
  const float inv = 1.0f / (float)(K11 * K11);

  // ---------- Phase A: async-load I,p halo tiles (zero-fill OOB) ----------
  for (int idx = tid; idx < SH * SW; idx += 256) {
    const int ly = idx / SW, lx = idx - ly * SW;
    const int gy = oy0 - 2 * RADIUS + ly;
    const int gx = ox0 - 2 * RADIUS + lx;
    if ((unsigned)gy < (unsigned)IMH && (unsigned)gx < (unsigned)IMW) {
      const long go = (long)gy * IMW + gx;
      async_cp_b32(gI + go, &lds[OFF_SI + idx]);
      async_cp_b32(gP + go, &lds[OFF_SP + idx]);
    } else {
      lds[OFF_SI + idx] = 0.0f;
      lds[OFF_SP + idx] = 0.0f;
    }
  }
  async_wait_all();
  __syncthreads();

  // ---------- Phase B: horizontal 11-sums of I, p, I*p, I*I (sliding) -----
  {
    const int SEG = (AW + 3) / 4;                    // 19
    for (int t = tid; t < SH * 4; t += 256) {        // 272 items
      const int row = t >> 2, seg = t & 3;
      const int cx0 = seg * SEG;
      const int cx1 = imin(AW, cx0 + SEG);
      const float* rI = &lds[OFF_SI + row * SW];
      const float* rP = &lds[OFF_SP + row * SW];
      float s1 = 0.f, s2 = 0.f, s3 = 0.f, s4 = 0.f;
#pragma unroll
      for (int k = 0; k < K11 - 1; ++k) {
        const float vi = rI[cx0 + k], vp = rP[cx0 + k];
        s1 += vi; s2 += vp; s3 += vi * vp; s4 += vi * vi;
      }
      for (int cx = cx0; cx < cx1; ++cx) {
        const float vi = rI[cx + K11 - 1], vp = rP[cx + K11 - 1];
        s1 += vi; s2 += vp; s3 += vi * vp; s4 += vi * vi;
        const int o = row * AW + cx;
        lds[OFF_HI  + o] = s1;
        lds[OFF_HP  + o] = s2;
        lds[OFF_HIP + o] = s3;
        lds[OFF_HII + o] = s4;
        const float ui = rI[cx], up = rP[cx];
        s1 -= ui; s2 -= up; s3 -= ui * up; s4 -= ui * ui;
      }
    }
  }
  __syncthreads();

  // ---------- Phase C: vertical 11-sums -> a, b (zero outside image) ------
  {
    const int SEG = (AH + 3) / 4;                    // 15
    for (int t = tid; t < AW * 4; t += 256) {        // 296 items
      const int ax = t >> 2, seg = t & 3;
      const int ay0 = seg * SEG;
      const int ay1 = imin(AH, ay0 + SEG);
      float s1 = 0.f, s2 = 0.f, s3 = 0.f, s4 = 0.f;
#pragma unroll
      for (int k = 0; k < K11 - 1; ++k) {
        const int o = (ay0 + k) * AW + ax;
        s1 += lds[OFF_HI + o]; s2 += lds[OFF_HP + o];
        s3 += lds[OFF_HIP + o]; s4 += lds[OFF_HII + o];
      }
      for (int ay = ay0; ay < ay1; ++ay) {
        const int oin = (ay + K11 - 1) * AW + ax;
        s1 += lds[OFF_HI + oin]; s2 += lds[OFF_HP + oin];
        s3 += lds[OFF_HIP + oin]; s4 += lds[OFF_HII + oin];
        const float mI  = s1 * inv, mP = s2 * inv;
        const float mIp = s3 * inv, mII = s4 * inv;
        const float a = (mIp - mI * mP) / (mII - mI * mI + EPSF);
        const float b = mP - a * mI;
        const int gy = oy0 - RADIUS + ay;
        const int gx = ox0 - RADIUS + ax;
        const bool in = ((unsigned)gy < (unsigned)IMH) &&
                        ((unsigned)gx < (unsigned)IMW);
        const int o = ay * AW + ax;
        lds[OFF_AT + o] = in ? a : 0.0f;   // box(a) is zero-padded in reference
        lds[OFF_BT + o] = in ? b : 0.0f;
        const int oout = ay * AW + ax;
        s1 -= lds[OFF_HI + oout]; s2 -= lds[OFF_HP + oout];
        s3 -= lds[OFF_HIP + oout]; s4 -= lds[OFF_HII + oout];
      }
    }
  }
  __syncthreads();

  // ---------- Phase D: horizontal 11-sums of a, b -------------------------
  {
    const int SEG = TW / 4;                          // 16
    for (int t = tid; t < AH * 4; t += 256) {        // 232 items
      const int row = t >> 2, seg = t & 3;
      const int x0 = seg * SEG, x1 = x0 + SEG;
      const float* ra = &lds[OFF_AT + row * AW];
      const float* rb = &lds[OFF_BT + row * AW];
      float sa = 0.f, sb = 0.f;
#pragma unroll
      for (int k = 0; k < K11 - 1; ++k) { sa += ra[x0 + k]; sb += rb[x0 + k]; }
      for (int ox = x0; ox < x1; ++ox) {
        sa += ra[ox + K11 - 1]; sb += rb[ox + K11 - 1];
        const int o = row * TW + ox;
        lds[OFF_HA + o] = sa;
        lds[OFF_HB + o] = sb;
        sa -= ra[ox]; sb -= rb[ox];
      }
    }
  }
  __syncthreads();

  // ---------- Phase E: vertical 11-sums -> q staged in LDS ---------------
  {
    const int SEG = TH / 4;                          // 12
    for (int t = tid; t < TW * 4; t += 256) {        // 256 items exactly
      const int ox = t >> 2, seg = t & 3;
      const int y0 = seg * SEG, y1 = y0 + SEG;
      float sa = 0.f, sb = 0.f;
#pragma unroll
      for (int k = 0; k < K11 - 1; ++k) {
        const int o = (y0 + k) * TW + ox;
        sa += lds[OFF_HA + o]; sb += lds[OFF_HB + o];
      }
      for (int oy = y0; oy < y1; ++oy) {
        const int oin = (oy + K11 - 1) * TW + ox;
        sa += lds[OFF_HA + oin]; sb += lds[OFF_HB + oin];
        const float Iv = lds[OFF_SI + (oy + 2 * RADIUS) * SW + (ox + 2 * RADIUS)];
        lds[OFF_QT + oy * TW + ox] = (sa * Iv + sb) * inv;
        const int oo = oy * TW + ox;
        sa -= lds[OFF_HA + oo]; sb -= lds[OFF_HB + oo];
      }
    }
  }
  __syncthreads();

  // ---------- Writeback: coalesced float4 stores -------------------------
  for (int idx = tid; idx < (TH * TW) / 4; idx += 256) {
    const int oy  = idx / (TW / 4);
    const int oxq = idx - oy * (TW / 4);
    const int gy  = oy0 + oy;
    if (gy < IMH) {
      const float4 v = *(const float4*)&lds[OFF_QT + oy * TW + oxq * 4];
      *(float4*)&Q[planeOff + (long)gy * IMW + ox0 + oxq * 4] = v;
    }
  }
}

extern "C" void kernel_launch(void* const* d_in, const int* in_sizes, int n_in,
                              void* d_out, int out_size, void* d_ws, size_t ws_size,
                              hipStream_t stream) {
  (void)in_sizes; (void)n_in; (void)out_size; (void)d_ws; (void)ws_size;
  const float* I = (const float*)d_in[0];
  const float* P = (const float*)d_in[1];
  float* Q = (float*)d_out;

  dim3 grid(IMW / TW, (IMH + TH - 1) / TH, BATCH * CHANS);  // 16 x 22 x 48
  dim3 block(256, 1, 1);                                    // 8 wave32
  guided_filter_fused<<<grid, block, 0, stream>>>(I, P, Q);
}